// ColorGNN_2843268350531
// MI455X (gfx1250) — compile-verified
//
#include <hip/hip_runtime.h>

typedef __attribute__((ext_vector_type(2))) float v2f;
typedef __attribute__((ext_vector_type(8))) float v8f;

#define B_ROWS 512
#define C_COLS 512
#define HID    128
#define NNODES 1024
#define NH     (NNODES * HID)   // 131072

// ---------------------------------------------------------------------------
// Per-row bitonic argsort (descending, stable -> smaller index first on ties)
// One block per row, 512 elements in LDS.
// ---------------------------------------------------------------------------
__global__ __launch_bounds__(256) void argsort_rows_desc(const float* __restrict__ probs,
                                                         int* __restrict__ idx) {
  __shared__ float sv[512];
  __shared__ int   si[512];
  const int row = blockIdx.x;
  const float* p = probs + row * 512;
  for (int t = threadIdx.x; t < 512; t += 256) { sv[t] = p[t]; si[t] = t; }
  __syncthreads();
  for (int k = 2; k <= 512; k <<= 1) {
    for (int j = k >> 1; j > 0; j >>= 1) {
      for (int t = threadIdx.x; t < 512; t += 256) {
        const int ixj = t ^ j;
        if (ixj > t) {
          const bool desc = ((t & k) == 0);
          float va = sv[t], vb = sv[ixj];
          int   ia = si[t], ib = si[ixj];
          const bool aAfterB = (va < vb) || (va == vb && ia > ib);
          if (desc ? aAfterB : !aAfterB) {
            sv[t] = vb; sv[ixj] = va; si[t] = ib; si[ixj] = ia;
          }
        }
      }
      __syncthreads();
    }
  }
  for (int t = threadIdx.x; t < 512; t += 256) idx[row * 512 + t] = si[t];
}

// ---------------------------------------------------------------------------
// f32 WMMA GEMM: D = A@B (+ A1@B1 if DUAL) + biasScale*bias, optional ReLU.
// A: MxK row-major; B: KxN row-major; D: MxN. N is a compile-time constant so
// all B-tile addressing strength-reduces to immediate offsets.
// Block = 256 threads (8 waves). Each wave owns a 32x16 output tile (two
// 16x16 f32 accumulators sharing every B fragment). K is consumed in chunks
// of 16 (4 x K4 steps): all fragment loads for a chunk are issued before the
// 8 (16 if DUAL) V_WMMA_F32_16X16X4_F32 ops, so one loadcnt wait covers the
// whole chunk instead of one wait per WMMA.
// If Bw_alt != null, row-groups with m0 >= M/2 use Bw_alt (bipartite halves).
// ---------------------------------------------------------------------------
template <int N, bool DUAL>
__global__ __launch_bounds__(256) void gemm_f32_wmma(
    const float* __restrict__ A, const float* __restrict__ Bw,
    const float* __restrict__ Bw_alt,
    const float* __restrict__ A1, const float* __restrict__ B1,
    const float* __restrict__ bias, float biasScale,
    float* __restrict__ D, int M, int K, int doRelu) {
  const int wave = threadIdx.x >> 5;
  const int lane = threadIdx.x & 31;
  const int half = lane >> 4;      // 0: lanes 0-15, 1: lanes 16-31
  const int l15  = lane & 15;
  const int m0   = blockIdx.x * 32;
  const int n0   = blockIdx.y * 128 + wave * 16;
  const int nc   = n0 + l15;

  const float* Bsel = (Bw_alt != nullptr && m0 >= (M >> 1)) ? Bw_alt : Bw;
  const float* Arow0 = A + (size_t)(m0 + l15) * K;
  const float* Arow1 = A + (size_t)(m0 + 16 + l15) * K;
  const float* A1row0 = DUAL ? (A1 + (size_t)(m0 + l15) * K) : nullptr;
  const float* A1row1 = DUAL ? (A1 + (size_t)(m0 + 16 + l15) * K) : nullptr;

  v8f acc0 = {};
  v8f acc1 = {};

  for (int k0 = 0; k0 < K; k0 += 16) {
    // Chunk bases: lane half selects K pair {0,1} vs {2,3} inside each K4 step.
    const float* Bp = Bsel + (size_t)(k0 + 2 * half) * N + nc;
    v2f a0[4], a1[4], bf[4];
#pragma unroll
    for (int s = 0; s < 4; ++s) {
      const int ka = k0 + 4 * s + 2 * half;
      a0[s].x = Arow0[ka];  a0[s].y = Arow0[ka + 1];
      a1[s].x = Arow1[ka];  a1[s].y = Arow1[ka + 1];
      bf[s].x = Bp[4 * s * N];  bf[s].y = Bp[4 * s * N + N];
    }
    v2f c0[4], c1[4], df[4];
    if constexpr (DUAL) {
      const float* B1p = B1 + (size_t)(k0 + 2 * half) * N + nc;
#pragma unroll
      for (int s = 0; s < 4; ++s) {
        const int ka = k0 + 4 * s + 2 * half;
        c0[s].x = A1row0[ka];  c0[s].y = A1row0[ka + 1];
        c1[s].x = A1row1[ka];  c1[s].y = A1row1[ka + 1];
        df[s].x = B1p[4 * s * N];  df[s].y = B1p[4 * s * N + N];
      }
    }
#pragma unroll
    for (int s = 0; s < 4; ++s) {
      acc0 = __builtin_amdgcn_wmma_f32_16x16x4_f32(false, a0[s], false, bf[s],
                                                   (short)0, acc0, false, false);
      acc1 = __builtin_amdgcn_wmma_f32_16x16x4_f32(false, a1[s], false, bf[s],
                                                   (short)0, acc1, false, false);
      if constexpr (DUAL) {
        acc0 = __builtin_amdgcn_wmma_f32_16x16x4_f32(false, c0[s], false, df[s],
                                                     (short)0, acc0, false, false);
        acc1 = __builtin_amdgcn_wmma_f32_16x16x4_f32(false, c1[s], false, df[s],
                                                     (short)0, acc1, false, false);
      }
    }
  }

  const float bv = (bias != nullptr) ? bias[nc] * biasScale : 0.0f;
#pragma unroll
  for (int v = 0; v < 8; ++v) {
    float v0 = acc0[v] + bv;
    float v1 = acc1[v] + bv;
    if (doRelu) { v0 = fmaxf(v0, 0.0f); v1 = fmaxf(v1, 0.0f); }
    D[(size_t)(m0 + v + 8 * half) * N + nc] = v0;
    D[(size_t)(m0 + 16 + v + 8 * half) * N + nc] = v1;
  }
}

// ---------------------------------------------------------------------------
// Pairwise relu partial reduction over a 128-wide j-slice:
//   Spart[p, side*512 + i, h] = sum_{j in slice p} max(U[i,h] + V[j,h] + b1[h], 0)
// blockIdx.x: [0,32) bird rows / [32,64) color rows (U,V swapped);
// blockIdx.y: j-slice p in [0,4). 128 threads = channel h; 16 rows in registers
// amortize each V load across 16 VALU ops. Partials are later reduced in a
// FIXED order (no float atomics) to keep the output bitwise deterministic.
// ---------------------------------------------------------------------------
__global__ __launch_bounds__(128) void pairwise_relu_sum_part(
    const float* __restrict__ UV, const float* __restrict__ b1,
    float* __restrict__ Spart) {
  const int h    = threadIdx.x;
  const int side = blockIdx.x >> 5;
  const int i0   = (blockIdx.x & 31) * 16;
  const int p    = blockIdx.y;
  const float* Up = UV + (size_t)side * (B_ROWS * HID);
  const float* Vp = UV + (size_t)(1 - side) * (B_ROWS * HID);
  const float bh = b1[h];
  float u[16], s[16];
#pragma unroll
  for (int r = 0; r < 16; ++r) { u[r] = Up[(i0 + r) * HID + h] + bh; s[r] = 0.0f; }
  const int jbeg = p * 128;
#pragma unroll 4
  for (int j = jbeg; j < jbeg + 128; ++j) {
    const float vj = Vp[j * HID + h];
#pragma unroll
    for (int r = 0; r < 16; ++r) s[r] += fmaxf(u[r] + vj, 0.0f);
  }
  float* out = Spart + (size_t)p * NH + (size_t)(side * B_ROWS + i0) * HID;
#pragma unroll
  for (int r = 0; r < 16; ++r) out[r * HID + h] = s[r];
}

// S = Spart[0] + Spart[1] + Spart[2] + Spart[3]  (fixed order -> deterministic)
__global__ __launch_bounds__(256) void reduce_spart(const float* __restrict__ Spart,
                                                    float* __restrict__ S) {
  const int t = blockIdx.x * 256 + threadIdx.x;
  S[t] = ((Spart[t] + Spart[NH + t]) + Spart[2 * NH + t]) + Spart[3 * NH + t];
}

// color_nodes = eye(C) @ W + b  ->  just W + b broadcast
__global__ __launch_bounds__(256) void color_nodes_init(const float* __restrict__ Wp,
                                                        const float* __restrict__ b,
                                                        float* __restrict__ xcol) {
  const int t = blockIdx.x * 256 + threadIdx.x;
  if (t < C_COLS * HID) xcol[t] = Wp[t] + b[t & (HID - 1)];
}

// cost[i,j] = 1 - gnn[i, idx[i,j]] * probs[i, idx[i,j]]   (mask == 1 everywhere)
__global__ __launch_bounds__(256) void finalize_cost(const float* __restrict__ gnn,
                                                     const float* __restrict__ probs,
                                                     const int* __restrict__ idx,
                                                     float* __restrict__ cost) {
  const int t = blockIdx.x * 256 + threadIdx.x;
  if (t < B_ROWS * C_COLS) {
    const int i = t >> 9;
    const int c = idx[t];
    cost[t] = 1.0f - gnn[i * 512 + c] * probs[i * 512 + c];
  }
}

extern "C" void kernel_launch(void* const* d_in, const int* in_sizes, int n_in,
                              void* d_out, int out_size, void* d_ws, size_t ws_size,
                              hipStream_t stream) {
  const float* probs  = (const float*)d_in[0];   // 512x512
  const float* Wp     = (const float*)d_in[1];   // 512x128
  const float* bp     = (const float*)d_in[2];   // 128
  const float* eW1    = (const float*)d_in[3];   // 3x256x128
  const float* eb1    = (const float*)d_in[4];   // 3x128
  const float* eW2    = (const float*)d_in[5];   // 3x128x128
  const float* eb2    = (const float*)d_in[6];   // 3x128
  const float* nW1    = (const float*)d_in[7];   // 3x256x128
  const float* nb1    = (const float*)d_in[8];   // 3x128
  const float* nW2    = (const float*)d_in[9];   // 3x128x128
  const float* nb2    = (const float*)d_in[10];  // 3x128
  const float* Wc     = (const float*)d_in[11];  // 128x512
  const float* cb     = (const float*)d_in[12];  // 512
  float* cost = (float*)d_out;                   // 512x512

  float* xA    = (float*)d_ws;
  float* xB    = xA + NH;
  float* UV    = xB + NH;        // U = UV[:512], V = UV[512:]
  float* S     = UV + NH;        // S_bird | S_color
  float* aggr  = S + NH;
  float* hid   = aggr + NH;
  float* Spart = hid + NH;       // 4 x NH partial sums
  float* gnn   = Spart + 4 * NH; // 512x512
  int*   idx   = (int*)(gnn + B_ROWS * C_COLS);

  // 1) idx = per-row descending argsort of probs (top_k with k == C)
  argsort_rows_desc<<<B_ROWS, 256, 0, stream>>>(probs, idx);

  // 2) x[:512] = probs @ Wp + bp  (filtered == probs since k == C);
  //    x[512:] = Wp + bp (identity color features)
  gemm_f32_wmma<HID, false><<<dim3(B_ROWS / 32, 1), 256, 0, stream>>>(
      probs, Wp, nullptr, nullptr, nullptr, bp, 1.0f, xA, B_ROWS, C_COLS, 0);
  color_nodes_init<<<(C_COLS * HID) / 256, 256, 0, stream>>>(Wp, bp, xA + B_ROWS * HID);

  float* x  = xA;
  float* xn = xB;
  for (int l = 0; l < 3; ++l) {
    const float* W1  = eW1 + (size_t)l * 2 * HID * HID;  // top: bird, bottom: color
    const float* b1  = eb1 + l * HID;
    const float* W2  = eW2 + (size_t)l * HID * HID;
    const float* b2  = eb2 + l * HID;
    const float* NW1 = nW1 + (size_t)l * 2 * HID * HID;
    const float* b1n = nb1 + l * HID;
    const float* NW2 = nW2 + (size_t)l * HID * HID;
    const float* b2n = nb2 + l * HID;

    // U = x_bird @ W1_top ; V = x_color @ W1_bot (weight selected per half)
    gemm_f32_wmma<HID, false><<<dim3(NNODES / 32, 1), 256, 0, stream>>>(
        x, W1, W1 + HID * HID, nullptr, nullptr, nullptr, 1.0f, UV, NNODES, HID, 0);

    // S[i] = sum_j relu(U[i]+V[j]+b1), both sides, j split 4-ways then reduced
    pairwise_relu_sum_part<<<dim3(64, 4), 128, 0, stream>>>(UV, b1, Spart);
    reduce_spart<<<NH / 256, 256, 0, stream>>>(Spart, S);

    // aggr = S @ W2 + 512*b2  (each node aggregates 512 edges)
    gemm_f32_wmma<HID, false><<<dim3(NNODES / 32, 1), 256, 0, stream>>>(
        S, W2, nullptr, nullptr, nullptr, b2, 512.0f, aggr, NNODES, HID, 0);

    // hid = relu(x @ NW1_top + aggr @ NW1_bot + b1n)
    gemm_f32_wmma<HID, true><<<dim3(NNODES / 32, 1), 256, 0, stream>>>(
        x, NW1, nullptr, aggr, NW1 + HID * HID, b1n, 1.0f, hid, NNODES, HID, 1);

    // x_new = hid @ NW2 + b2n
    gemm_f32_wmma<HID, false><<<dim3(NNODES / 32, 1), 256, 0, stream>>>(
        hid, NW2, nullptr, nullptr, nullptr, b2n, 1.0f, xn, NNODES, HID, 0);

    float* t = x; x = xn; xn = t;
  }

  // gnn = x[:512] @ Wc + cb   (M=512, K=128, N=512)
  gemm_f32_wmma<C_COLS, false><<<dim3(B_ROWS / 32, C_COLS / 128), 256, 0, stream>>>(
      x, Wc, nullptr, nullptr, nullptr, cb, 1.0f, gnn, B_ROWS, HID, 0);

  // cost = 1 - gather(gnn * probs, idx)
  finalize_cost<<<(B_ROWS * C_COLS) / 256, 256, 0, stream>>>(gnn, probs, idx, cost);
}